// DetNMSPostProcessor_48627619726095
// MI455X (gfx1250) — compile-verified
//
#include <hip/hip_runtime.h>
#include <hip/hip_bf16.h>
#include <math.h>

// ---------------- problem constants ----------------
#define Bb 4
#define Nn 300
#define Cc 80
#define Kk 17
#define Ww 64
#define HWsz 4096
#define IOU_THR 0.7f
#define SCORE_THR 0.01f
#define NEGV (-1000000000.0f)
#define NEGINF (-3.402823466e38f)

// workspace layout (in 4-byte elements)
#define WS_MAXC   0      // 4 uints: per-batch max |coord| (atomicMax on float bits)
#define WS_SCORES 4      // 4*300 floats
#define WS_LABELS 1204   // 4*300 ints
#define WS_BOXES  2404   // 4*300*4 floats (16B aligned: 2404*4 = 9616)
#define WS_KPTS   7204   // 4*300*17*3 floats

// output layout (floats): fl | fb | fs | akpts | vkeep
#define O_FL 0
#define O_FB 1200
#define O_FS 6000
#define O_FK 7200
#define O_FV 68400

// NMS block config
#define NP    304        // 300 padded to 19*16
#define NTI   19
#define WORDS 10         // ceil(304/32)
#define SORTN 512

typedef __attribute__((ext_vector_type(2))) float v2f;
typedef __attribute__((ext_vector_type(8))) float v8f;
typedef __attribute__((ext_vector_type(4))) int   v4i;

typedef __attribute__((address_space(1))) v4i as1_v4i;
typedef __attribute__((address_space(3))) v4i as3_v4i;

#if __has_builtin(__builtin_amdgcn_wmma_f32_16x16x4_f32)
#define HAVE_WMMA4 1
#endif
#if __has_builtin(__builtin_amdgcn_global_load_async_to_lds_b128) && __has_builtin(__builtin_amdgcn_s_wait_asynccnt)
#define HAVE_ASYNC 1
#endif

// ---------------- kernel 0: init maxc accumulators ----------------
__global__ void init_kernel(float* ws) {
    if (threadIdx.x < Bb) ((unsigned*)ws)[WS_MAXC + threadIdx.x] = 0u;
}

// ---------------- kernel 1: scores / labels / abs boxes ----------------
__global__ void score_box_kernel(const float* __restrict__ logits,
                                 const float* __restrict__ boxes,
                                 const float* __restrict__ osz,
                                 float* __restrict__ ws) {
    int i = blockIdx.x * blockDim.x + threadIdx.x;
    if (i >= Bb * Nn) return;
    int b = i / Nn;

    // argmax over C=80 logits, first-occurrence
    const float* lg = logits + (size_t)i * Cc;
    float m = NEGINF; int am = 0;
    for (int c = 0; c < Cc; ++c) {
        float v = lg[c];
        if (v > m) { m = v; am = c; }
    }
    float score = 1.0f / (1.0f + expf(-m));

    // cxcywh -> xyxy, scale by orig_target_sizes (tiled [s0,s1,s0,s1])
    const float* bx = boxes + (size_t)i * 4;
    float cx = bx[0], cy = bx[1], w = bx[2], h = bx[3];
    float sx = osz[b * 2 + 0], sy = osz[b * 2 + 1];
    float x1 = (cx - 0.5f * w) * sx;
    float y1 = (cy - 0.5f * h) * sy;
    float x2 = (cx + 0.5f * w) * sx;
    float y2 = (cy + 0.5f * h) * sy;

    ws[WS_SCORES + i] = score;
    ((int*)ws)[WS_LABELS + i] = am;
    float* ob = ws + WS_BOXES + (size_t)i * 4;
    ob[0] = x1; ob[1] = y1; ob[2] = x2; ob[3] = y2;

    // per-batch max |coord| for class-separation offset (values >= 0 -> float bits monotonic)
    float amax = fmaxf(fmaxf(fabsf(x1), fabsf(y1)), fmaxf(fabsf(x2), fabsf(y2)));
    atomicMax(((unsigned*)ws) + WS_MAXC + b, __float_as_uint(amax));
}

// ---------------- kernel 2: heatmap max/argmax (bandwidth-bound) ----------------
// one wave32 per (b,n,k) tile of 4096 f32; coalesced b128 loads, shuffle reduce
__global__ __launch_bounds__(256) void heatmap_kernel(const float* __restrict__ heat,
                                                      const float* __restrict__ koff,
                                                      float* __restrict__ ws) {
    int gwave = blockIdx.x * (blockDim.x >> 5) + (threadIdx.x >> 5);
    if (gwave >= Bb * Nn * Kk) return;
    int lane = threadIdx.x & 31;

    const float4* base = (const float4*)(heat + (size_t)gwave * HWsz);
    float bestV = NEGINF; int bestI = 0;
    #pragma unroll 4
    for (int it = 0; it < 32; ++it) {
        float4 v = base[it * 32 + lane];
        int i0 = (it * 32 + lane) * 4;
        if (v.x > bestV) { bestV = v.x; bestI = i0;     }
        if (v.y > bestV) { bestV = v.y; bestI = i0 + 1; }
        if (v.z > bestV) { bestV = v.z; bestI = i0 + 2; }
        if (v.w > bestV) { bestV = v.w; bestI = i0 + 3; }
    }
    // wave reduce, tie -> lowest index (jnp.argmax first occurrence)
    for (int off = 16; off > 0; off >>= 1) {
        float ov = __shfl_xor(bestV, off);
        int   oi = __shfl_xor(bestI, off);
        if (ov > bestV || (ov == bestV && oi < bestI)) { bestV = ov; bestI = oi; }
    }
    if (lane == 0) {
        const float* of = koff + (size_t)gwave * 2;
        float nx = (float)(bestI & 63) * (1.0f / 63.0f) + of[0];
        float ny = (float)(bestI >> 6) * (1.0f / 63.0f) + of[1];
        nx = fminf(fmaxf(nx, 0.0f), 1.0f);
        ny = fminf(fmaxf(ny, 0.0f), 1.0f);
        float* o = ws + WS_KPTS + (size_t)gwave * 3;
        o[0] = nx; o[1] = ny; o[2] = bestV;
    }
}

// ---------------- kernel 3: per-batch sort + IoU(WMMA) + greedy NMS + output ----------------
__global__ __launch_bounds__(SORTN) void nms_kernel(const float* __restrict__ ws,
                                                    float* __restrict__ out) {
    __shared__ float    s_key[SORTN];
    __shared__ int      s_idx[SORTN];
    __shared__ __align__(16) float s_rawbox[Nn * 4];
    __shared__ int      s_rawlab[Nn];
    __shared__ float    s_ob[NP * 4];   // class-offset boxes, sorted order
    __shared__ float    s_bx[NP * 4];   // plain abs boxes, sorted order
    __shared__ float    s_area[NP];
    __shared__ int      s_lab[NP];
    __shared__ unsigned s_mask[NP * WORDS];
    __shared__ int      s_keep[NP];

    const int b = blockIdx.x;
    const int t = threadIdx.x;
    const int lane = t & 31;

    const float* scores = ws + WS_SCORES + b * Nn;
    const int*   labels = ((const int*)ws) + WS_LABELS + b * Nn;
    const float* boxes  = ws + WS_BOXES + (size_t)b * Nn * 4;

    // ---- stage raw boxes/labels into LDS (CDNA5 async-to-LDS path when available) ----
#ifdef HAVE_ASYNC
    if (t < Nn) {
        __builtin_amdgcn_global_load_async_to_lds_b128(
            (as1_v4i*)(boxes + t * 4),
            (as3_v4i*)(&s_rawbox[t * 4]), 0, 0);
        s_rawlab[t] = labels[t];
    }
    __builtin_amdgcn_s_wait_asynccnt(0);
#else
    if (t < Nn) {
        ((float4*)s_rawbox)[t] = ((const float4*)boxes)[t];
        s_rawlab[t] = labels[t];
    }
#endif

    // ---- sort keys: s = valid ? score : NEG ; pads -> -inf ----
    if (t < Nn) {
        float s = scores[t];
        s_key[t] = (s > SCORE_THR) ? s : NEGV;
        s_idx[t] = t;
    } else {
        s_key[t] = NEGINF;
        s_idx[t] = t;
    }
    __syncthreads();

    // ---- bitonic sort, descending by key, ties -> lower original index (stable argsort) ----
    for (int k = 2; k <= SORTN; k <<= 1) {
        for (int j = k >> 1; j > 0; j >>= 1) {
            int ixj = t ^ j;
            if (ixj > t) {
                float ka = s_key[t], kb2 = s_key[ixj];
                int ia = s_idx[t], ib = s_idx[ixj];
                bool aFirst = (ka > kb2) || (ka == kb2 && ia < ib);
                bool dirDesc = ((t & k) == 0);
                bool doSwap = dirDesc ? !aFirst : aFirst;
                if (doSwap) {
                    s_key[t] = kb2; s_key[ixj] = ka;
                    s_idx[t] = ib;  s_idx[ixj] = ia;
                }
            }
            __syncthreads();
        }
    }

    // ---- gather sorted boxes, apply class offset, compute areas; clear masks ----
    float maxc = __uint_as_float(((const unsigned*)ws)[WS_MAXC + b]) + 1.0f;
    if (t < NP) {
        if (t < Nn) {
            int o = s_idx[t];
            float x1 = s_rawbox[o * 4 + 0], y1 = s_rawbox[o * 4 + 1];
            float x2 = s_rawbox[o * 4 + 2], y2 = s_rawbox[o * 4 + 3];
            int lb = s_rawlab[o];
            float ofs = (float)lb * maxc;
            s_bx[t * 4 + 0] = x1; s_bx[t * 4 + 1] = y1;
            s_bx[t * 4 + 2] = x2; s_bx[t * 4 + 3] = y2;
            s_ob[t * 4 + 0] = x1 + ofs; s_ob[t * 4 + 1] = y1 + ofs;
            s_ob[t * 4 + 2] = x2 + ofs; s_ob[t * 4 + 3] = y2 + ofs;
            s_area[t] = (x2 - x1) * (y2 - y1);
            s_lab[t] = lb;
        } else {
            s_bx[t * 4 + 0] = 0.f; s_bx[t * 4 + 1] = 0.f; s_bx[t * 4 + 2] = 0.f; s_bx[t * 4 + 3] = 0.f;
            s_ob[t * 4 + 0] = 0.f; s_ob[t * 4 + 1] = 0.f; s_ob[t * 4 + 2] = 0.f; s_ob[t * 4 + 3] = 0.f;
            s_area[t] = 0.f;
            s_lab[t] = 0;
        }
        s_keep[t] = 0;
    }
    for (int w = t; w < NP * WORDS; w += blockDim.x) s_mask[w] = 0u;
    __syncthreads();

    // ---- IoU suppression bitmasks over 16x16 tiles ----
    // denominator area_i + area_j is a rank-2 outer sum -> v_wmma_f32_16x16x4_f32
    {
        const int wv = t >> 5;
        const int nw = blockDim.x >> 5;
        for (int tile = wv; tile < NTI * NTI; tile += nw) {
            int ti = tile / NTI, tj = tile % NTI;
            int i0 = ti * 16, j0 = tj * 16;
            float den8[8];
#ifdef HAVE_WMMA4
            // A(16x4): row m = [area_i0+m, 1, 0, 0]; B(4x16): col n = [1; area_j0+n; 0; 0]
            v2f av, bv;
            if (lane < 16) {
                av.x = s_area[i0 + lane]; av.y = 1.0f;   // K=0, K=1 rows (lanes 0-15)
                bv.x = 1.0f;              bv.y = s_area[j0 + lane];
            } else {
                av.x = 0.0f; av.y = 0.0f;                // K=2, K=3 contribute zero
                bv.x = 0.0f; bv.y = 0.0f;
            }
            v8f cacc = {};
            cacc = __builtin_amdgcn_wmma_f32_16x16x4_f32(
                false, av, false, bv, (short)0, cacc, false, false);
            #pragma unroll
            for (int e = 0; e < 8; ++e) den8[e] = cacc[e];
#else
            #pragma unroll
            for (int e = 0; e < 8; ++e) {
                int M = e + ((lane >= 16) ? 8 : 0);
                den8[e] = s_area[i0 + M] + s_area[j0 + (lane & 15)];
            }
#endif
            #pragma unroll
            for (int e = 0; e < 8; ++e) {
                int M = e + ((lane >= 16) ? 8 : 0);
                int i = i0 + M;
                int j = j0 + (lane & 15);
                if (i < Nn && j < Nn && j > i) {
                    float ix1 = fmaxf(s_ob[i * 4 + 0], s_ob[j * 4 + 0]);
                    float iy1 = fmaxf(s_ob[i * 4 + 1], s_ob[j * 4 + 1]);
                    float ix2 = fminf(s_ob[i * 4 + 2], s_ob[j * 4 + 2]);
                    float iy2 = fminf(s_ob[i * 4 + 3], s_ob[j * 4 + 3]);
                    float iw = fmaxf(ix2 - ix1, 0.0f);
                    float ih = fmaxf(iy2 - iy1, 0.0f);
                    float inter = iw * ih;
                    float iou = inter / (den8[e] - inter + 1e-9f);
                    if (iou > IOU_THR)
                        atomicOr(&s_mask[i * WORDS + (j >> 5)], 1u << (j & 31));
                }
            }
        }
    }
    __syncthreads();

    // ---- sequential greedy NMS: one wave, 300-bit supp mask lives in 10 lanes ----
    if (t < 32) {
        unsigned suppw = 0u;
        for (int i = 0; i < Nn; ++i) {
            unsigned wsel = __shfl(suppw, i >> 5);
            bool suppressed = ((wsel >> (i & 31)) & 1u) != 0u;
            bool vs = s_key[i] > -9.0e8f;            // NEG entries are invalid
            bool kp = vs && !suppressed;
            if (lane == 0) s_keep[i] = kp ? 1 : 0;
            if (kp && lane < WORDS) suppw |= s_mask[i * WORDS + lane];
        }
    }
    __syncthreads();

    // ---- top_k reorder (kept first in sorted order, then non-kept by index) + outputs ----
    if (t < Nn) {
        int before = 0, total = 0;
        for (int j2 = 0; j2 < Nn; ++j2) {
            int kj = s_keep[j2];
            total += kj;
            if (j2 < t) before += kj;
        }
        int kpi = s_keep[t];
        int p = kpi ? before : (total + t - before);
        int sel = s_idx[t];
        int op = b * Nn + p;

        float x1 = s_bx[t * 4 + 0], y1 = s_bx[t * 4 + 1];
        float x2 = s_bx[t * 4 + 2], y2 = s_bx[t * 4 + 3];

        out[O_FL + op] = kpi ? (float)s_lab[t] : -1.0f;
        out[O_FB + op * 4 + 0] = kpi ? x1 : 0.0f;
        out[O_FB + op * 4 + 1] = kpi ? y1 : 0.0f;
        out[O_FB + op * 4 + 2] = kpi ? x2 : 0.0f;
        out[O_FB + op * 4 + 3] = kpi ? y2 : 0.0f;
        out[O_FS + op] = kpi ? s_key[t] : 0.0f;
        out[O_FV + op] = kpi ? 1.0f : 0.0f;

        const float* kp3 = ws + WS_KPTS + ((size_t)(b * Nn + sel)) * Kk * 3;
        for (int kk = 0; kk < Kk; ++kk) {
            float nx = kp3[kk * 3 + 0], ny = kp3[kk * 3 + 1], mv = kp3[kk * 3 + 2];
            float ax = x1 + nx * (x2 - x1);
            float ay = y1 + ny * (y2 - y1);
            size_t ko = O_FK + ((size_t)op * Kk + kk) * 3;
            out[ko + 0] = kpi ? ax : 0.0f;
            out[ko + 1] = kpi ? ay : 0.0f;
            out[ko + 2] = kpi ? mv : 0.0f;
        }
    }
}

// ---------------- launcher ----------------
extern "C" void kernel_launch(void* const* d_in, const int* in_sizes, int n_in,
                              void* d_out, int out_size, void* d_ws, size_t ws_size,
                              hipStream_t stream) {
    const float* logits = (const float*)d_in[0];   // (4,300,80)
    const float* boxes  = (const float*)d_in[1];   // (4,300,4)
    const float* heat   = (const float*)d_in[2];   // (4,300,17,64,64)
    const float* koff   = (const float*)d_in[3];   // (4,300,17,2)
    const float* osz    = (const float*)d_in[4];   // (4,2)
    float* ws  = (float*)d_ws;
    float* out = (float*)d_out;

    init_kernel<<<1, 64, 0, stream>>>(ws);
    score_box_kernel<<<(Bb * Nn + 255) / 256, 256, 0, stream>>>(logits, boxes, osz, ws);
    {
        int waves = Bb * Nn * Kk;                 // 20400 tiles
        int wpb = 256 / 32;                       // 8 waves per block
        heatmap_kernel<<<(waves + wpb - 1) / wpb, 256, 0, stream>>>(heat, koff, ws);
    }
    nms_kernel<<<Bb, SORTN, 0, stream>>>(ws, out);
}